// RGCN_66125316489905
// MI455X (gfx1250) — compile-verified
//
#include <hip/hip_runtime.h>
#include <hip/hip_bf16.h>

typedef float v2f __attribute__((ext_vector_type(2)));
typedef float v8f __attribute__((ext_vector_type(8)));

#define N_NODES 100000
#define FDIM 128
#define OUT_STRIDE 512

// ---------------- utility kernels ----------------

__global__ void zero_f32_kernel(float* __restrict__ p, int n) {
    int i = blockIdx.x * blockDim.x + threadIdx.x;
    if (i < n) p[i] = 0.0f;
}

__global__ void degree_kernel(const int* __restrict__ src, const int* __restrict__ dst,
                              float* __restrict__ deg_out, float* __restrict__ deg_in, int E) {
    int e = blockIdx.x * blockDim.x + threadIdx.x;
    if (e >= E) return;
    atomicAdd(deg_out + src[e], 1.0f);
    atomicAdd(deg_in + dst[e], 1.0f);
}

__global__ void invsqrt_kernel(float* __restrict__ d, int n) {
    int i = blockIdx.x * blockDim.x + threadIdx.x;
    if (i >= n) return;
    d[i] = rsqrtf(fmaxf(d[i], 1.0f));   // DGL clamps degree to >= 1
}

// copy feat into out[:, 0:128] (skip connection 0), float4 vectorized
__global__ void copy_feat_kernel(const float* __restrict__ feat, float* __restrict__ out) {
    int idx = blockIdx.x * blockDim.x + threadIdx.x;      // N_NODES * 32 threads
    if (idx >= N_NODES * (FDIM / 4)) return;
    int i  = idx >> 5;          // node
    int c4 = idx & 31;          // float4 column
    float4 v = ((const float4*)(feat + (size_t)i * FDIM))[c4];
    ((float4*)(out + (size_t)i * OUT_STRIDE))[c4] = v;
}

// ---------------- FP32 WMMA GEMM: tmp = (h * inv_sqrt_out[:,None]) @ W ----------------
// Block = 256 threads = 8 waves. Block covers 16 rows; wave w covers cols [16w, 16w+16).
// One V_WMMA_F32_16X16X4_F32 per K-step of 4; K = 128 -> 32 WMMAs per wave.
__global__ void __launch_bounds__(256)
gemm_wmma_f32_kernel(const float* __restrict__ H, int ldh,
                     const float* __restrict__ W,          // 128x128 row-major
                     const float* __restrict__ inv_out,
                     float* __restrict__ out)              // N x 128, row-major
{
    const int wave = threadIdx.x >> 5;
    const int lane = threadIdx.x & 31;
    const int row0 = blockIdx.x * 16;
    const int col0 = wave * 16;

    const int m    = lane & 15;       // A-matrix row / B,C,D column within tile
    const int half = lane >> 4;       // 0: lanes 0-15, 1: lanes 16-31
    const int kb   = half * 2;        // A/B K sub-offset for this half-wave

    const int row = row0 + m;
    const float s = inv_out[row];                       // fold degree scale into A
    const float* __restrict__ hrow = H + (size_t)row * ldh;

    v8f c = {};
    #pragma unroll 4
    for (int k0 = 0; k0 < FDIM; k0 += 4) {
        v2f a, b;
        a.x = hrow[k0 + kb]     * s;                    // A[m][k0+kb]
        a.y = hrow[k0 + kb + 1] * s;                    // A[m][k0+kb+1]
        b.x = W[(size_t)(k0 + kb)     * FDIM + col0 + m];  // B[k][n]
        b.y = W[(size_t)(k0 + kb + 1) * FDIM + col0 + m];
        c = __builtin_amdgcn_wmma_f32_16x16x4_f32(
                /*neg_a=*/false, a, /*neg_b=*/false, b,
                /*c_mod=*/(short)0, c, /*reuse_a=*/false, /*reuse_b=*/false);
    }

    // C/D layout: VGPR r holds row m = r + 8*half, col n = lane&15
    #pragma unroll
    for (int r = 0; r < 8; ++r) {
        out[(size_t)(row0 + half * 8 + r) * FDIM + col0 + m] = c[r];
    }
}

// ---------------- SpMM scatter-add: agg[dst[e],:] += tmp[src[e],:] ----------------
// One wave32 per edge; each lane moves one float4 (coalesced 512B gather per edge),
// accumulation via global f32 atomics (L2-resident: 51.2MB feature matrix << 192MB L2).
__global__ void spmm_kernel(const float* __restrict__ tmp,
                            const int* __restrict__ src, const int* __restrict__ dst,
                            float* __restrict__ agg, int E) {
    int gid  = blockIdx.x * blockDim.x + threadIdx.x;
    int e    = gid >> 5;
    int lane = threadIdx.x & 31;
    if (e >= E) return;
    int si = src[e];
    int di = dst[e];
    float4 v = ((const float4*)(tmp + (size_t)si * FDIM))[lane];
    float* o = agg + (size_t)di * FDIM + lane * 4;
    atomicAdd(o + 0, v.x);
    atomicAdd(o + 1, v.y);
    atomicAdd(o + 2, v.z);
    atomicAdd(o + 3, v.w);
}

// ---------------- finalize: out_section = relu(agg * inv_sqrt_in[:,None] + b) ----------------
__global__ void finalize_kernel(const float* __restrict__ agg,
                                const float* __restrict__ inv_in,
                                const float* __restrict__ bias,
                                float* __restrict__ outsec) {   // points at out + (l+1)*128
    int idx = blockIdx.x * blockDim.x + threadIdx.x;            // N_NODES * 32 threads
    if (idx >= N_NODES * (FDIM / 4)) return;
    int i  = idx >> 5;
    int c4 = idx & 31;
    float s = inv_in[i];
    float4 v = ((const float4*)(agg + (size_t)i * FDIM))[c4];
    float4 b = ((const float4*)bias)[c4];
    float4 r;
    r.x = fmaxf(v.x * s + b.x, 0.0f);
    r.y = fmaxf(v.y * s + b.y, 0.0f);
    r.z = fmaxf(v.z * s + b.z, 0.0f);
    r.w = fmaxf(v.w * s + b.w, 0.0f);
    ((float4*)(outsec + (size_t)i * OUT_STRIDE))[c4] = r;
}

// ---------------- launch ----------------

extern "C" void kernel_launch(void* const* d_in, const int* in_sizes, int n_in,
                              void* d_out, int out_size, void* d_ws, size_t ws_size,
                              hipStream_t stream) {
    const float* feat = (const float*)d_in[0];
    const float* Wl[3] = {(const float*)d_in[1], (const float*)d_in[3], (const float*)d_in[5]};
    const float* bl[3] = {(const float*)d_in[2], (const float*)d_in[4], (const float*)d_in[6]};
    const int* src = (const int*)d_in[7];
    const int* dst = (const int*)d_in[8];
    const int E = in_sizes[7];
    const int N = N_NODES;

    float* out = (float*)d_out;
    float* ws  = (float*)d_ws;

    float* inv_out = ws;                        // N floats (degree, then rsqrt in place)
    float* inv_in  = ws + N;                    // N floats
    float* tmp     = ws + 2 * (size_t)N;        // N*128 floats
    float* agg     = tmp + (size_t)N * FDIM;    // N*128 floats

    const int T = 256;

    // degrees -> inv-sqrt (clamped)
    zero_f32_kernel<<<(2 * N + T - 1) / T, T, 0, stream>>>(inv_out, 2 * N);
    degree_kernel<<<(E + T - 1) / T, T, 0, stream>>>(src, dst, inv_out, inv_in, E);
    invsqrt_kernel<<<(2 * N + T - 1) / T, T, 0, stream>>>(inv_out, 2 * N);

    // skip 0: out[:, 0:128] = feat
    copy_feat_kernel<<<(N * (FDIM / 4) + T - 1) / T, T, 0, stream>>>(feat, out);

    const int gemm_blocks = N / 16;                 // 100000 / 16 = 6250, exact
    const int feat_threads = N * (FDIM / 4);        // elementwise float4 kernels
    const int spmm_threads_blocks = (E * 32 + T - 1) / T;

    for (int l = 0; l < 3; ++l) {
        const float* h;
        int ldh;
        if (l == 0) { h = feat;                ldh = FDIM; }
        else        { h = out + l * FDIM;      ldh = OUT_STRIDE; }  // previous skip section

        gemm_wmma_f32_kernel<<<gemm_blocks, T, 0, stream>>>(h, ldh, Wl[l], inv_out, tmp);

        zero_f32_kernel<<<((size_t)N * FDIM + T - 1) / T, T, 0, stream>>>(agg, N * FDIM);

        spmm_kernel<<<spmm_threads_blocks, T, 0, stream>>>(tmp, src, dst, agg, E);

        finalize_kernel<<<(feat_threads + T - 1) / T, T, 0, stream>>>(
            agg, inv_in, bl[l], out + (l + 1) * FDIM);
    }
}